// LGGSALayer_16389595201618
// MI455X (gfx1250) — compile-verified
//
#include <hip/hip_runtime.h>
#include <math.h>

// ---------------------------------------------------------------------------
// CDNA5 / gfx1250 implementation of the LGGSA layer, fp32 WMMA (16x16x4).
// ---------------------------------------------------------------------------

typedef float v2f __attribute__((ext_vector_type(2)));
typedef float v8f __attribute__((ext_vector_type(8)));

static __device__ __forceinline__ v8f wmma4(v2f a, v2f b, v8f c) {
  // D = A(16x4) * B(4x16) + C(16x16), fp32, wave32.
  return __builtin_amdgcn_wmma_f32_16x16x4_f32(false, a, false, b, (short)0, c,
                                               false, false);
}

// A-fragment: lane m (0..15) holds row m; lanes 16..31 hold K=+2,+3.
static __device__ __forceinline__ v2f ldA(const float* base, int ld, int k0) {
  int l = threadIdx.x & 31;
  int m = l & 15;
  int kh = (l >> 4) << 1;
  v2f a;
  a.x = base[m * ld + k0 + kh];
  a.y = base[m * ld + k0 + kh + 1];
  return a;
}

// B-fragment: lane n (0..15) holds column n0+n; lanes 16..31 hold K=+2,+3.
static __device__ __forceinline__ v2f ldB(const float* base, int ld, int k0,
                                          int n0) {
  int l = threadIdx.x & 31;
  int n = l & 15;
  int kh = (l >> 4) << 1;
  v2f b;
  b.x = base[(k0 + kh) * ld + n0 + n];
  b.y = base[(k0 + kh + 1) * ld + n0 + n];
  return b;
}

// D-fragment store: vgpr j -> row (lane<16 ? j : j+8), col = lane&15.
static __device__ __forceinline__ void stD(float* base, int ld, int n0, v8f c) {
  int l = threadIdx.x & 31;
  int n = l & 15;
  int mb = (l >> 4) << 3;
#pragma unroll
  for (int j = 0; j < 8; ++j) base[(mb + j) * ld + n0 + n] = c[j];
}

// ---------------------------------------------------------------------------
// Kernel A: per-patch local self-attention + detail pooling.
// One wave per 16-pixel patch, 2 waves per block.  32768 patches.
// ---------------------------------------------------------------------------
__global__ __launch_bounds__(64) void lsa_patch_kernel(
    const float* __restrict__ x, const float* __restrict__ Wq,
    const float* __restrict__ bq, const float* __restrict__ Wo,
    const float* __restrict__ bo, const float* __restrict__ dlW,
    const float* __restrict__ dlb, float* __restrict__ lsa_out,
    float* __restrict__ y_out) {
  __shared__ __align__(16) float wq_s[64 * 64];  // [k][n]
  __shared__ __align__(16) float wo_s[64 * 64];  // [k][n]
  __shared__ __align__(16) float dlw_s[64 * 16]; // [d][n]
  __shared__ float bq_s[64], bo_s[64], dlb_s[16];
  __shared__ __align__(16) float xs[2][16 * 68];
  __shared__ __align__(16) float qs[2][16 * 68];
  __shared__ float ss[2][16 * 17];
  __shared__ float misc[2][96];  // [0:64) mean, [64:80) logits, [80:96) probs

  int tid = threadIdx.x;
  for (int i = tid; i < 64 * 64; i += 64) {
    wq_s[i] = Wq[i];
    wo_s[i] = Wo[i];
  }
  for (int i = tid; i < 64 * 16; i += 64) dlw_s[i] = dlW[i];
  if (tid < 64) {
    bq_s[tid] = bq[tid];
    bo_s[tid] = bo[tid];
  }
  if (tid < 16) dlb_s[tid] = dlb[tid];
  __syncthreads();

  int wid = tid >> 5, lane = tid & 31;
  int pid = blockIdx.x * 2 + wid;  // 0..32767
  int b = pid >> 14;
  int rem = pid & 16383;
  int pr = rem >> 7, pc = rem & 127;
  const float* xbase = x + ((size_t)(b * 512 + pr * 4) * 512 + pc * 4) * 64;
  float* xw = xs[wid];
  float* qw = qs[wid];
  float* sw = ss[wid];
  float* mw = misc[wid];

  // load 16x64 X tile (patch pixels) into LDS
#pragma unroll
  for (int it = 0; it < 8; ++it) {
    int t = lane + it * 32;
    int px = t >> 4, c4 = t & 15;
    int pi = px >> 2, pj = px & 3;
    float4 v = *(const float4*)(xbase + (size_t)(pi * 512 + pj) * 64 + c4 * 4);
    *(float4*)(xw + px * 68 + c4 * 4) = v;
  }
  __syncthreads();

  // q = X @ Wq + bq
#pragma unroll
  for (int nt = 0; nt < 4; ++nt) {
    v8f c = {};
#pragma unroll
    for (int k0 = 0; k0 < 64; k0 += 4)
      c = wmma4(ldA(xw, 68, k0), ldB(wq_s, 64, k0, nt * 16), c);
    float bias = bq_s[nt * 16 + (lane & 15)];
#pragma unroll
    for (int j = 0; j < 8; ++j) c[j] += bias;
    stD(qw, 68, nt * 16, c);
  }
  __syncthreads();

  // S = q @ q^T  (16x16)
  {
    v8f c = {};
#pragma unroll
    for (int k0 = 0; k0 < 64; k0 += 4) {
      v2f a = ldA(qw, 68, k0);
      int n = lane & 15, kh = (lane >> 4) << 1;
      v2f bb;
      bb.x = qw[n * 68 + k0 + kh];      // B[k][n] = q[n][k]
      bb.y = qw[n * 68 + k0 + kh + 1];
      c = wmma4(a, bb, c);
    }
    stD(sw, 17, 0, c);
  }
  __syncthreads();
  if (lane < 16) {  // row softmax
    float* rr = sw + lane * 17;
    float mx = rr[0];
    for (int j = 1; j < 16; ++j) mx = fmaxf(mx, rr[j]);
    float sum = 0.f;
    for (int j = 0; j < 16; ++j) {
      float e = expf(rr[j] - mx);
      rr[j] = e;
      sum += e;
    }
    float inv = 1.f / sum;
    for (int j = 0; j < 16; ++j) rr[j] *= inv;
  }
  __syncthreads();

  // lsa_pre = attn @ q  -> xw (reuse)
#pragma unroll
  for (int nt = 0; nt < 4; ++nt) {
    v8f c = {};
#pragma unroll
    for (int k0 = 0; k0 < 16; k0 += 4)
      c = wmma4(ldA(sw, 17, k0), ldB(qw, 68, k0, nt * 16), c);
    stD(xw, 68, nt * 16, c);
  }
  __syncthreads();

  // lsa = lsa_pre @ Wo + bo -> qw (reuse)
#pragma unroll
  for (int nt = 0; nt < 4; ++nt) {
    v8f c = {};
#pragma unroll
    for (int k0 = 0; k0 < 64; k0 += 4)
      c = wmma4(ldA(xw, 68, k0), ldB(wo_s, 64, k0, nt * 16), c);
    float bias = bo_s[nt * 16 + (lane & 15)];
#pragma unroll
    for (int j = 0; j < 8; ++j) c[j] += bias;
    stD(qw, 68, nt * 16, c);
  }
  __syncthreads();

  // write lsa tile to global (patch layout)
  float* lbase = lsa_out + (size_t)pid * 16 * 64;
#pragma unroll
  for (int it = 0; it < 8; ++it) {
    int t = lane + it * 32;
    int px = t >> 4, c4 = t & 15;
    *(float4*)(lbase + px * 64 + c4 * 4) =
        *(const float4*)(qw + px * 68 + c4 * 4);
  }

  // mean over pixels
  {
    float s0 = 0.f, s1 = 0.f;
    for (int i2 = 0; i2 < 16; ++i2) {
      s0 += qw[i2 * 68 + lane];
      s1 += qw[i2 * 68 + lane + 32];
    }
    mw[lane] = s0 * (1.f / 16.f);
    mw[lane + 32] = s1 * (1.f / 16.f);
  }
  __syncthreads();
  if (lane < 16) {  // logits = mean @ dl_W + dl_b
    float lg = dlb_s[lane];
    for (int d = 0; d < 64; ++d) lg += mw[d] * dlw_s[d * 16 + lane];
    mw[64 + lane] = lg;
  }
  __syncthreads();
  if (lane < 16) {  // softmax over 16 logits
    float mx = mw[64];
    for (int j = 1; j < 16; ++j) mx = fmaxf(mx, mw[64 + j]);
    float sum = 0.f;
    for (int j = 0; j < 16; ++j) sum += expf(mw[64 + j] - mx);
    mw[80 + lane] = expf(mw[64 + lane] - mx) / sum;
  }
  __syncthreads();
  // y = sum_i lsa[i] * prob[i]
  {
    float a0 = 0.f, a1 = 0.f;
    for (int i2 = 0; i2 < 16; ++i2) {
      float p = mw[80 + i2];
      a0 += qw[i2 * 68 + lane] * p;
      a1 += qw[i2 * 68 + lane + 32] * p;
    }
    y_out[(size_t)pid * 64 + lane] = a0;
    y_out[(size_t)pid * 64 + lane + 32] = a1;
  }
}

// ---------------------------------------------------------------------------
// Generic (Mx64) @ (64x64) + bias.  1 wave per 16-row tile, 4 waves/block.
// ---------------------------------------------------------------------------
__global__ __launch_bounds__(128) void gemm64_kernel(
    const float* __restrict__ A, const float* __restrict__ W,
    const float* __restrict__ bias, float* __restrict__ out) {
  __shared__ __align__(16) float ws_s[64 * 64];
  __shared__ float bs_s[64];
  __shared__ __align__(16) float as_[4][16 * 68];
  int tid = threadIdx.x;
  for (int i = tid; i < 4096; i += 128) ws_s[i] = W[i];
  if (tid < 64) bs_s[tid] = bias[tid];
  int wid = tid >> 5, lane = tid & 31;
  int tile = blockIdx.x * 4 + wid;
  const float* abase = A + (size_t)tile * 16 * 64;
  float* aw = as_[wid];
#pragma unroll
  for (int it = 0; it < 8; ++it) {
    int t = lane + it * 32;
    int r = t >> 4, c4 = t & 15;
    *(float4*)(aw + r * 68 + c4 * 4) = *(const float4*)(abase + r * 64 + c4 * 4);
  }
  __syncthreads();
  float* obase = out + (size_t)tile * 16 * 64;
  int n = lane & 15, mb = (lane >> 4) << 3;
#pragma unroll
  for (int nt = 0; nt < 4; ++nt) {
    v8f c = {};
#pragma unroll
    for (int k0 = 0; k0 < 64; k0 += 4)
      c = wmma4(ldA(aw, 68, k0), ldB(ws_s, 64, k0, nt * 16), c);
    float bv = bs_s[nt * 16 + n];
#pragma unroll
    for (int j = 0; j < 8; ++j) obase[(mb + j) * 64 + nt * 16 + n] = c[j] + bv;
  }
}

// ---------------------------------------------------------------------------
// Axial (row / col) attention with Gaussian distance bias.
// One block per (b, line).  mode 0: rows (write), mode 1: cols (accumulate).
// ---------------------------------------------------------------------------
__global__ __launch_bounds__(64) void axial_attn_kernel(
    const float* __restrict__ q, const float* __restrict__ kk,
    const float* __restrict__ v, float* __restrict__ out,
    const float* __restrict__ sigma, int mode) {
  __shared__ __align__(16) float kt_s[64 * 128];   // K^T: [channel][pos]
  __shared__ __align__(16) float ps[2][16 * 132];  // per-wave prob tile
  int tid = threadIdx.x;
  int b = blockIdx.x >> 7, line = blockIdx.x & 127;
  int base, rstride;
  if (mode == 0) {
    base = ((b * 128 + line) * 128) * 64;
    rstride = 64;
  } else {
    base = ((b * 128) * 128 + line) * 64;
    rstride = 128 * 64;
  }
  float sv = sigma[0];
  float gw = 1.0f / (2.0f * sv * sv);

  for (int xp = tid; xp < 128; xp += 64) {
    const float* src = kk + (size_t)base + (size_t)xp * rstride;
    for (int c = 0; c < 64; ++c) kt_s[c * 128 + xp] = src[c];
  }
  __syncthreads();

  int wid = tid >> 5, lane = tid & 31;
  int n = lane & 15, mb = (lane >> 4) << 3;
  float* pw = ps[wid];

  for (int mi = 0; mi < 4; ++mi) {
    int i0 = (wid * 4 + mi) * 16;
    const float* qbase = q + (size_t)base + (size_t)i0 * rstride;
    // scores S = Q K^T - dist^2 * gw
#pragma unroll
    for (int xt = 0; xt < 8; ++xt) {
      v8f c = {};
#pragma unroll
      for (int k0 = 0; k0 < 64; k0 += 4)
        c = wmma4(ldA(qbase, rstride, k0), ldB(kt_s, 128, k0, xt * 16), c);
#pragma unroll
      for (int j = 0; j < 8; ++j) {
        float di = (float)((i0 + mb + j) - (xt * 16 + n));
        pw[(mb + j) * 132 + xt * 16 + n] = c[j] - di * di * gw;
      }
    }
    __syncthreads();
    if (lane < 16) {  // row softmax over 128
      float* rr = pw + lane * 132;
      float mx = rr[0];
      for (int j = 1; j < 128; ++j) mx = fmaxf(mx, rr[j]);
      float sum = 0.f;
      for (int j = 0; j < 128; ++j) {
        float e = expf(rr[j] - mx);
        rr[j] = e;
        sum += e;
      }
      float inv = 1.f / sum;
      for (int j = 0; j < 128; ++j) rr[j] *= inv;
    }
    __syncthreads();
    // out_tile = P @ V
    const float* vbase = v + (size_t)base;
#pragma unroll
    for (int nt = 0; nt < 4; ++nt) {
      v8f c = {};
#pragma unroll
      for (int k0 = 0; k0 < 128; k0 += 4)
        c = wmma4(ldA(pw, 132, k0), ldB(vbase, rstride, k0, nt * 16), c);
#pragma unroll
      for (int j = 0; j < 8; ++j) {
        size_t idx =
            (size_t)base + (size_t)(i0 + mb + j) * rstride + nt * 16 + n;
        if (mode == 0)
          out[idx] = c[j];
        else
          out[idx] += c[j];
      }
    }
    __syncthreads();
  }
}

// ---------------------------------------------------------------------------
// Final: out = concat(lsa_img, bilinear_up(gsa)) @ sq_W + sq_b.
// One wave per 16-pixel tile (same image row), 2 waves/block.
// ---------------------------------------------------------------------------
__global__ __launch_bounds__(64) void fuse_out_kernel(
    const float* __restrict__ lsa, const float* __restrict__ gsa,
    const float* __restrict__ sqW, const float* __restrict__ sqb,
    float* __restrict__ out) {
  __shared__ __align__(16) float ws_s[128 * 64];
  __shared__ float bs_s[64];
  __shared__ __align__(16) float as_[2][16 * 132];
  int tid = threadIdx.x;
  for (int i = tid; i < 128 * 64; i += 64) ws_s[i] = sqW[i];
  if (tid < 64) bs_s[tid] = sqb[tid];

  int wid = tid >> 5, lane = tid & 31;
  int tile = blockIdx.x * 2 + wid;  // 0..32767
  int gr = tile >> 5;               // b*512 + i
  int j0 = (tile & 31) * 16;
  int b = gr >> 9, ii = gr & 511;
  float* aw = as_[wid];
  int m = lane >> 1, d0 = (lane & 1) * 32;
  int jj = j0 + m;

  // lsa half (cols 0..63) from patch layout
  {
    int pr = ii >> 2, pi = ii & 3, pc = jj >> 2, pj = jj & 3;
    const float* src =
        lsa + ((size_t)((b * 128 + pr) * 128 + pc) * 16 + pi * 4 + pj) * 64 + d0;
    float* dst = aw + m * 132 + d0;
#pragma unroll
    for (int t = 0; t < 8; ++t)
      *(float4*)(dst + t * 4) = *(const float4*)(src + t * 4);
  }
  // gsa half (cols 64..127): align-corners bilinear 128 -> 512 (h then w)
  {
    const float S = 127.0f / 511.0f;
    float ph = ii * S;
    int h0 = (int)ph;
    if (h0 > 126) h0 = 126;
    float fh = ph - (float)h0;
    float pw = jj * S;
    int w0 = (int)pw;
    if (w0 > 126) w0 = 126;
    float fw = pw - (float)w0;
    const float* g00 = gsa + ((size_t)(b * 128 + h0) * 128 + w0) * 64 + d0;
    const float* g01 = g00 + 64;
    const float* g10 = g00 + 128 * 64;
    const float* g11 = g10 + 64;
    float* dst = aw + m * 132 + 64 + d0;
    float ofh = 1.f - fh, ofw = 1.f - fw;
#pragma unroll
    for (int t = 0; t < 8; ++t) {
      float4 a00 = *(const float4*)(g00 + t * 4);
      float4 a01 = *(const float4*)(g01 + t * 4);
      float4 a10 = *(const float4*)(g10 + t * 4);
      float4 a11 = *(const float4*)(g11 + t * 4);
      float4 r;
      r.x = (a00.x * ofh + a10.x * fh) * ofw + (a01.x * ofh + a11.x * fh) * fw;
      r.y = (a00.y * ofh + a10.y * fh) * ofw + (a01.y * ofh + a11.y * fh) * fw;
      r.z = (a00.z * ofh + a10.z * fh) * ofw + (a01.z * ofh + a11.z * fh) * fw;
      r.w = (a00.w * ofh + a10.w * fh) * ofw + (a01.w * ofh + a11.w * fh) * fw;
      *(float4*)(dst + t * 4) = r;
    }
  }
  __syncthreads();

  int n = lane & 15, mb = (lane >> 4) << 3;
#pragma unroll
  for (int nt = 0; nt < 4; ++nt) {
    v8f c = {};
#pragma unroll
    for (int k0 = 0; k0 < 128; k0 += 4)
      c = wmma4(ldA(aw, 132, k0), ldB(ws_s, 64, k0, nt * 16), c);
    float bv = bs_s[nt * 16 + n];
#pragma unroll
    for (int j = 0; j < 8; ++j)
      out[((size_t)gr * 512 + j0 + mb + j) * 64 + nt * 16 + n] = c[j] + bv;
  }
}

// ---------------------------------------------------------------------------
extern "C" void kernel_launch(void* const* d_in, const int* in_sizes, int n_in,
                              void* d_out, int out_size, void* d_ws,
                              size_t ws_size, hipStream_t stream) {
  (void)in_sizes;
  (void)n_in;
  (void)out_size;
  (void)ws_size;
  const float* x = (const float*)d_in[0];
  const float* lsa_Wq = (const float*)d_in[1];
  const float* lsa_bq = (const float*)d_in[2];
  const float* lsa_Wo = (const float*)d_in[3];
  const float* lsa_bo = (const float*)d_in[4];
  const float* dl_W = (const float*)d_in[5];
  const float* dl_b = (const float*)d_in[6];
  const float* gsa_Wq = (const float*)d_in[7];
  const float* gsa_bq = (const float*)d_in[8];
  const float* gsa_Wk = (const float*)d_in[9];
  const float* gsa_bk = (const float*)d_in[10];
  const float* gsa_Wv = (const float*)d_in[11];
  const float* gsa_bv = (const float*)d_in[12];
  const float* gsa_Wo = (const float*)d_in[13];
  const float* gsa_bo = (const float*)d_in[14];
  const float* sigma = (const float*)d_in[15];
  const float* sq_W = (const float*)d_in[16];
  const float* sq_b = (const float*)d_in[17];

  float* ws = (float*)d_ws;
  const size_t N_LSA = (size_t)2 * 128 * 128 * 16 * 64;  // 33,554,432
  const size_t N_Y = (size_t)2 * 128 * 128 * 64;         //  2,097,152
  float* lsa_buf = ws;
  float* y_buf = lsa_buf + N_LSA;
  float* gq = y_buf + N_Y;
  float* gk = gq + N_Y;
  float* gv = gk + N_Y;
  float* rc = gv + N_Y;
  float* gsab = rc + N_Y;

  lsa_patch_kernel<<<16384, 64, 0, stream>>>(x, lsa_Wq, lsa_bq, lsa_Wo, lsa_bo,
                                             dl_W, dl_b, lsa_buf, y_buf);
  gemm64_kernel<<<512, 128, 0, stream>>>(y_buf, gsa_Wq, gsa_bq, gq);
  gemm64_kernel<<<512, 128, 0, stream>>>(y_buf, gsa_Wk, gsa_bk, gk);
  gemm64_kernel<<<512, 128, 0, stream>>>(y_buf, gsa_Wv, gsa_bv, gv);
  axial_attn_kernel<<<256, 64, 0, stream>>>(gq, gk, gv, rc, sigma, 0);
  axial_attn_kernel<<<256, 64, 0, stream>>>(gq, gk, gv, rc, sigma, 1);
  gemm64_kernel<<<512, 128, 0, stream>>>(rc, gsa_Wo, gsa_bo, gsab);
  fuse_out_kernel<<<16384, 64, 0, stream>>>(lsa_buf, gsab, sq_W, sq_b,
                                            (float*)d_out);
}